// Net_33294586479043
// MI455X (gfx1250) — compile-verified
//
#include <hip/hip_runtime.h>
#include <hip/hip_bf16.h>

// ---------------------------------------------------------------------------
// GCN(1 -> 64 -> 1) on N=100000 nodes, E=1.6M edges.
// Width-1 input/output lets both propagation layers collapse to SCALAR
// per-edge messages (L2-resident fp32 atomics); the only dense math is
// relu(s*W1+b1) @ W2 per 16-node tile, done with V_WMMA_F32_16X16X4_F32.
// ---------------------------------------------------------------------------

typedef __attribute__((ext_vector_type(2))) float v2f;
typedef __attribute__((ext_vector_type(8))) float v8f;

#define TPB 256

// --- zero the three atomic accumulators (deg, s1, s2) every launch ----------
__global__ void zero_acc_kernel(float* __restrict__ deg,
                                float* __restrict__ s1,
                                float* __restrict__ s2, int n) {
  int i = blockIdx.x * blockDim.x + threadIdx.x;
  if (i < n) { deg[i] = 0.0f; s1[i] = 0.0f; s2[i] = 0.0f; }
}

// --- in-degree via fp32 atomics (counts are exact in fp32 here) -------------
__global__ void degree_kernel(const int* __restrict__ col,
                              float* __restrict__ deg, int E) {
  int i = blockIdx.x * blockDim.x + threadIdx.x;
  int stride = gridDim.x * blockDim.x;
  int n4 = E >> 2;
  const int4* col4 = reinterpret_cast<const int4*>(col);
  for (int e = i; e < n4; e += stride) {
    __builtin_prefetch(&col4[e + stride], 0, 1);  // global_prefetch_b8
    int4 cc = col4[e];
    atomicAdd(&deg[cc.x], 1.0f);
    atomicAdd(&deg[cc.y], 1.0f);
    atomicAdd(&deg[cc.z], 1.0f);
    atomicAdd(&deg[cc.w], 1.0f);
  }
  for (int e = (n4 << 2) + i; e < E; e += stride)
    atomicAdd(&deg[col[e]], 1.0f);
}

// --- dinv = (deg + selfloop)^-1/2  (in place over deg) ----------------------
__global__ void dinv_kernel(float* __restrict__ deg, int n) {
  int i = blockIdx.x * blockDim.x + threadIdx.x;
  if (i < n) deg[i] = rsqrtf(deg[i] + 1.0f);
}

// --- scalar edge scatter: acc[col] += dinv[row] * src[row] ------------------
__global__ void edge_scatter_kernel(const int* __restrict__ row,
                                    const int* __restrict__ col,
                                    const float* __restrict__ dinv,
                                    const float* __restrict__ src,
                                    float* __restrict__ acc, int E) {
  int i = blockIdx.x * blockDim.x + threadIdx.x;
  int stride = gridDim.x * blockDim.x;
  int n4 = E >> 2;
  const int4* row4 = reinterpret_cast<const int4*>(row);
  const int4* col4 = reinterpret_cast<const int4*>(col);
  for (int e = i; e < n4; e += stride) {
    __builtin_prefetch(&row4[e + stride], 0, 1);
    __builtin_prefetch(&col4[e + stride], 0, 1);
    int4 rr = row4[e];
    int4 cc = col4[e];
    atomicAdd(&acc[cc.x], dinv[rr.x] * src[rr.x]);
    atomicAdd(&acc[cc.y], dinv[rr.y] * src[rr.y]);
    atomicAdd(&acc[cc.z], dinv[rr.z] * src[rr.z]);
    atomicAdd(&acc[cc.w], dinv[rr.w] * src[rr.w]);
  }
  for (int e = (n4 << 2) + i; e < E; e += stride) {
    int r = row[e], c = col[e];
    atomicAdd(&acc[c], dinv[r] * src[r]);
  }
}

// --- s[c] = dinv[c] * (s1raw[c] + dinv[c]*x[c])  (adds self-loop term) ------
__global__ void normalize_s_kernel(float* __restrict__ s1,
                                   const float* __restrict__ dinv,
                                   const float* __restrict__ x, int n) {
  int i = blockIdx.x * blockDim.x + threadIdx.x;
  if (i < n) {
    float d = dinv[i];
    s1[i] = d * (s1[i] + d * x[i]);
  }
}

// --- z[c] = relu(s[c]*W1 + b1) @ W2 via chained V_WMMA_F32_16X16X4_F32 ------
// One wave32 handles 16 nodes. A(16x4 f32): lanes 0-15 carry K={0,1},
// lanes 16-31 carry K={2,3} (ISA 7.12.2). B is W2[k] broadcast over all 16
// columns, so D[m][n] = z[m] for every n: lane 0 VGPRs = z[base..base+7],
// lane 16 VGPRs = z[base+8..base+15].
__global__ __launch_bounds__(TPB) void zlayer_wmma_kernel(
    const float* __restrict__ s, const float* __restrict__ W1,
    const float* __restrict__ b1, const float* __restrict__ W2,
    float* __restrict__ z, int n_nodes) {
  const int lane = threadIdx.x & 31;
  const int waveId = (blockIdx.x * blockDim.x + threadIdx.x) >> 5;
  const int base = waveId * 16;
  if (base >= n_nodes) return;  // uniform per wave: full wave exits

  const int m = lane & 15;
  const int ko = (lane >> 4) << 1;  // 0 for lanes 0-15, 2 for lanes 16-31
  int node = base + m;
  if (node >= n_nodes) node = n_nodes - 1;  // clamp reads, keep EXEC all-1s
  const float sm = s[node];

  v8f c = {};
#pragma unroll
  for (int kk = 0; kk < 16; ++kk) {
    const int k0 = kk * 4 + ko;
    v2f a, b;
    a.x = fmaxf(fmaf(sm, W1[k0],     b1[k0]),     0.0f);  // fused ReLU
    a.y = fmaxf(fmaf(sm, W1[k0 + 1], b1[k0 + 1]), 0.0f);
    b.x = W2[k0];
    b.y = W2[k0 + 1];
    // (neg_a, A, neg_b, B, c_mod, C, reuse_a, reuse_b)
    c = __builtin_amdgcn_wmma_f32_16x16x4_f32(false, a, false, b,
                                              (short)0, c, false, false);
  }

  if (lane == 0) {
#pragma unroll
    for (int i = 0; i < 8; ++i)
      if (base + i < n_nodes) z[base + i] = c[i];
  } else if (lane == 16) {
#pragma unroll
    for (int i = 0; i < 8; ++i)
      if (base + 8 + i < n_nodes) z[base + 8 + i] = c[i];
  }
}

// --- out[c] = v - log(exp(v)),  v = dinv[c]*(s2[c] + dinv[c]*z[c]) + b2 -----
// (log_softmax over a singleton axis; values are O(1) so exp never overflows)
__global__ void out_kernel(const float* __restrict__ s2,
                           const float* __restrict__ dinv,
                           const float* __restrict__ z,
                           const float* __restrict__ b2,
                           float* __restrict__ out, int n) {
  int i = blockIdx.x * blockDim.x + threadIdx.x;
  if (i < n) {
    float d = dinv[i];
    float v = d * (s2[i] + d * z[i]) + b2[0];
    out[i] = v - logf(expf(v));
  }
}

extern "C" void kernel_launch(void* const* d_in, const int* in_sizes, int n_in,
                              void* d_out, int out_size, void* d_ws, size_t ws_size,
                              hipStream_t stream) {
  const float* x    = (const float*)d_in[0];   // [N,1]
  const int*   ei   = (const int*)d_in[1];     // [2,E] flat: rows then cols
  const float* W1   = (const float*)d_in[2];   // [1,64]
  const float* b1   = (const float*)d_in[3];   // [64]
  const float* W2   = (const float*)d_in[4];   // [64,1]
  const float* b2   = (const float*)d_in[5];   // [1]
  float*       out  = (float*)d_out;           // [N,1]

  const int N = in_sizes[0];
  const int E = in_sizes[1] / 2;
  const int* row = ei;
  const int* col = ei + E;

  // workspace: 4 * N floats
  float* dinv = (float*)d_ws;      // deg accumulator, then deg^-1/2
  float* s1   = dinv + N;          // layer-1 scatter acc, then s
  float* z    = s1 + N;            // hidden @ W2 per node
  float* s2   = z + N;             // layer-2 scatter acc

  const int gN = (N + TPB - 1) / TPB;
  const int n4 = E >> 2;
  const int gE = (n4 + TPB - 1) / TPB;
  const int waves = (N + 15) / 16;
  const int gW = (waves * 32 + TPB - 1) / TPB;

  zero_acc_kernel<<<gN, TPB, 0, stream>>>(dinv, s1, s2, N);
  degree_kernel<<<gE, TPB, 0, stream>>>(col, dinv, E);
  dinv_kernel<<<gN, TPB, 0, stream>>>(dinv, N);
  edge_scatter_kernel<<<gE, TPB, 0, stream>>>(row, col, dinv, x, s1, E);
  normalize_s_kernel<<<gN, TPB, 0, stream>>>(s1, dinv, x, N);
  zlayer_wmma_kernel<<<gW, TPB, 0, stream>>>(s1, W1, b1, W2, z, N);
  edge_scatter_kernel<<<gE, TPB, 0, stream>>>(row, col, dinv, z, s2, E);
  out_kernel<<<gN, TPB, 0, stream>>>(s2, dinv, z, b2, out, N);
}